// Model_1657857377053
// MI455X (gfx1250) — compile-verified
//
#include <hip/hip_runtime.h>
#include <math.h>

typedef __attribute__((ext_vector_type(2))) float v2f;
typedef __attribute__((ext_vector_type(4))) float v4f;
typedef __attribute__((ext_vector_type(8))) float v8f;

// Quantum quanvolution collapsed to: per patch p[16] (real product state),
//   Re = (p*cos(phi)) @ H16 ; Im = (p*sin(phi)) @ H16  (H16[r][c] = 0.25*(-1)^popc(r&c))
//   probs = Re^2+Im^2 ; logits[c] = b[c] + sum_patch sum_s probs*G[c][patch][s]
// One wave handles a tile of 16 patches = 4 batch elements using
// V_WMMA_F32_16X16X4_F32 (fp32-exact, 4 k-steps x {Re,Im} = 8 WMMAs/tile).
__global__ __launch_bounds__(256) void quanv_softmax_kernel(
    const float* __restrict__ x,   // [B,4,4]
    const float* __restrict__ wq,  // [4] CRZ angles
    const float* __restrict__ Wc,  // [2,16]
    const float* __restrict__ bc,  // [2]
    float* __restrict__ out,       // [B,2]
    int Bn)
{
    const int lane  = threadIdx.x & 31;
    const int col   = lane & 15;   // B/C/D column (output state); also A row m (patch in tile)
    const int khalf = lane >> 4;   // which K-pair / which C-row half this lane holds

    // ---- per-lane constants (computed once) ----
    // CRZ diagonal phases for this lane's 8 K-slots: k = 4t + 2*khalf + j
    float cphi[8], sphi[8];
    {
        const float w0 = wq[0], w1 = wq[1], w2 = wq[2], w3 = wq[3];
#pragma unroll
        for (int t = 0; t < 4; ++t) {
#pragma unroll
            for (int j = 0; j < 2; ++j) {
                const int idx = 2 * t + j;
                const int k = 4 * t + 2 * khalf + j;
                const float a = (float)((k >> 3) & 1);
                const float b = (float)((k >> 2) & 1);
                const float c = (float)((k >> 1) & 1);
                const float d = (float)(k & 1);
                const float expo = 0.5f * (w0 * (2.f * b - 1.f) * a +
                                           w1 * (2.f * c - 1.f) * b +
                                           w2 * (2.f * d - 1.f) * c +
                                           w3 * (2.f * a - 1.f) * d);
                cphi[idx] = cosf(expo);
                sphi[idx] = sinf(expo);
            }
        }
    }

    // B operand: rows 4t+2*khalf(+1) of H16 at column `col` (constant, shared by Re/Im)
    v2f bmat[4];
#pragma unroll
    for (int t = 0; t < 4; ++t) {
        const int r0 = 4 * t + 2 * khalf;
        bmat[t].x = ((__popc(r0 & col) & 1) ? -0.25f : 0.25f);
        bmat[t].y = ((__popc((r0 + 1) & col) & 1) ? -0.25f : 0.25f);
    }

    // G[c][p] at state `col`: fold PauliZ signs into classifier weights
    float G0[4], G1[4];
#pragma unroll
    for (int p = 0; p < 4; ++p) {
        float g0 = 0.f, g1 = 0.f;
#pragma unroll
        for (int q = 0; q < 4; ++q) {
            const float sg = ((col >> (3 - q)) & 1) ? -1.f : 1.f;
            g0 += sg * Wc[4 * p + q];
            g1 += sg * Wc[16 + 4 * p + q];
        }
        G0[p] = g0;
        G1[p] = g1;
    }
    const float bias0 = bc[0], bias1 = bc[1];

    // A-row of this lane: patch m = col within the tile of 16
    const int lb   = col >> 2;       // local batch element (0..3)
    const int pId  = col & 3;        // patch within batch element
    const int jrow = pId >> 1, kcol = pId & 1;

    const int ntiles = (Bn + 3) >> 2;
    const int nwave  = (int)((gridDim.x * blockDim.x) >> 5);
    const int wid    = (int)((blockIdx.x * blockDim.x + threadIdx.x) >> 5);

    for (int tile = wid; tile < ntiles; tile += nwave) {
        const int nb = tile * 4;

        // prefetch next tile's 256B slab of x (uniform cond; speculative ok)
        const int nxt = tile + nwave;
        if (nxt < ntiles)
            __builtin_prefetch(x + (size_t)nxt * 64 + lane * 2, 0, 0);

        int nA = nb + lb;
        if (nA >= Bn) nA = Bn - 1;   // clamp tail (keeps EXEC full for WMMA)

        // patch features: x[n, 2j, 2k..2k+1] and x[n, 2j+1, 2k..2k+1]
        const float* xp = x + (size_t)nA * 16 + jrow * 8 + kcol * 2;
        const v2f lo = *(const v2f*)xp;
        const v2f hi = *(const v2f*)(xp + 4);

        const float c0 = __cosf(0.5f * lo.x), s0 = __sinf(0.5f * lo.x);
        const float c1 = __cosf(0.5f * lo.y), s1 = __sinf(0.5f * lo.y);
        const float c2 = __cosf(0.5f * hi.x), s2 = __sinf(0.5f * hi.x);
        const float c3 = __cosf(0.5f * hi.y), s3 = __sinf(0.5f * hi.y);

        // product-state partials: p[k] = e01[k>>2] * e23[k&3]; here k&3 = 2*khalf+j
        const float e01_0 = c0 * c1, e01_1 = c0 * s1, e01_2 = s0 * c1, e01_3 = s0 * s1;
        const float g2 = khalf ? s2 : c2;
        const float f0 = g2 * c3;   // e23[2*khalf]
        const float f1 = g2 * s3;   // e23[2*khalf+1]
        const float e01[4] = {e01_0, e01_1, e01_2, e01_3};

        v8f accRe = {};
        v8f accIm = {};
#pragma unroll
        for (int t = 0; t < 4; ++t) {
            const float p0 = e01[t] * f0;   // p[4t+2*khalf]
            const float p1 = e01[t] * f1;   // p[4t+2*khalf+1]
            v2f aRe, aIm;
            aRe.x = p0 * cphi[2 * t];     aRe.y = p1 * cphi[2 * t + 1];
            aIm.x = p0 * sphi[2 * t];     aIm.y = p1 * sphi[2 * t + 1];
            accRe = __builtin_amdgcn_wmma_f32_16x16x4_f32(
                false, aRe, false, bmat[t], (short)0, accRe, false, false);
            accIm = __builtin_amdgcn_wmma_f32_16x16x4_f32(
                false, aIm, false, bmat[t], (short)0, accIm, false, false);
        }

        // probs -> logits partials. C/D row r maps to patch m_row = 8*khalf + r:
        // r<4 => local batch 2*khalf, r>=4 => local batch 2*khalf+1
        float l00 = 0.f, l01 = 0.f, l10 = 0.f, l11 = 0.f;
#pragma unroll
        for (int r = 0; r < 8; ++r) {
            const float re = accRe[r], im = accIm[r];
            const float pr = fmaf(re, re, im * im);
            const int pp = r & 3;
            if (r < 4) {
                l00 = fmaf(pr, G0[pp], l00);
                l01 = fmaf(pr, G1[pp], l01);
            } else {
                l10 = fmaf(pr, G0[pp], l10);
                l11 = fmaf(pr, G1[pp], l11);
            }
        }

        // butterfly-reduce over the 16 lanes of each half (masks 1..8 stay in-half)
#pragma unroll
        for (int mk = 1; mk < 16; mk <<= 1) {
            l00 += __shfl_xor(l00, mk, 32);
            l01 += __shfl_xor(l01, mk, 32);
            l10 += __shfl_xor(l10, mk, 32);
            l11 += __shfl_xor(l11, mk, 32);
        }

        if (col == 0) {
            const int bA = nb + 2 * khalf;          // this half's first batch element
            const float eA = __expf((l01 + bias1) - (l00 + bias0));
            const float oA0 = 1.f / (1.f + eA);
            const float eB = __expf((l11 + bias1) - (l10 + bias0));
            const float oB0 = 1.f / (1.f + eB);
            if (bA + 1 < Bn) {
                v4f o = {oA0, 1.f - oA0, oB0, 1.f - oB0};
                *(v4f*)(out + (size_t)bA * 2) = o;
            } else if (bA < Bn) {
                out[(size_t)bA * 2 + 0] = oA0;
                out[(size_t)bA * 2 + 1] = 1.f - oA0;
            }
        }
    }
}

extern "C" void kernel_launch(void* const* d_in, const int* in_sizes, int n_in,
                              void* d_out, int out_size, void* d_ws, size_t ws_size,
                              hipStream_t stream) {
    const float* x  = (const float*)d_in[0];
    const float* wq = (const float*)d_in[1];
    const float* Wc = (const float*)d_in[2];
    const float* bc = (const float*)d_in[3];
    float* out = (float*)d_out;

    const int Bn = in_sizes[0] / 16;          // batch size
    const int ntiles = (Bn + 3) / 4;          // 16 patches (= 4 batch elems) per wave-tile
    const int threads = 256;                  // 8 waves per block
    int blocks = (ntiles + 7) / 8;
    if (blocks > 4096) blocks = 4096;
    if (blocks < 1) blocks = 1;

    quanv_softmax_kernel<<<blocks, threads, 0, stream>>>(x, wq, Wc, bc, out, Bn);
}